// Sparsemax_80487687127239
// MI455X (gfx1250) — compile-verified
//
#include <hip/hip_runtime.h>
#include <hip/hip_bf16.h>

#ifndef __has_builtin
#define __has_builtin(x) 0
#endif

#define D    2048   // row length (matches reference last axis)
#define NT   256    // threads per block (8 wave32)
#define NW   (NT / 32)
#define RPB  8      // rows processed per block (software pipelined)

typedef float v4f __attribute__((ext_vector_type(4)));
typedef int   v4i __attribute__((ext_vector_type(4)));
typedef __attribute__((address_space(1))) v4i gv4i;   // global int4
typedef __attribute__((address_space(3))) v4i lv4i;   // LDS int4
typedef __attribute__((address_space(3))) void as3_void;

#if __has_builtin(__builtin_amdgcn_global_load_async_to_lds_b128)
#define HAVE_ASYNC_B128 1
#endif
#if __has_builtin(__builtin_amdgcn_s_wait_asynccnt)
#define HAVE_WAIT_ASYNC 1
#endif

// Per-lane async copy of 16B from global to LDS (tracked by ASYNCcnt).
__device__ __forceinline__ void async_copy_b128(const float* gsrc, float* ldst) {
#if defined(HAVE_ASYNC_B128) && defined(__AMDGCN__)
    __builtin_amdgcn_global_load_async_to_lds_b128(
        (gv4i*)gsrc, (lv4i*)ldst, /*imm offset*/0, /*cpol*/0);
#else
    unsigned loff = (unsigned)(unsigned long long)(as3_void*)ldst;
    asm volatile("global_load_async_to_lds_b128 %0, %1, off"
                 :: "v"(loff), "v"(gsrc) : "memory");
#endif
}

#if defined(HAVE_WAIT_ASYNC) && defined(__AMDGCN__)
#define WAIT_ASYNCCNT(n) __builtin_amdgcn_s_wait_asynccnt(n)
#else
#define WAIT_ASYNCCNT(n) asm volatile("s_wait_asynccnt " #n ::: "memory")
#endif

// Block-wide max; 2 barriers; result broadcast to all threads.
__device__ __forceinline__ float block_reduce_max(float m, float* sh) {
#pragma unroll
    for (int i = 16; i > 0; i >>= 1) m = fmaxf(m, __shfl_xor(m, i, 32));
    const int wave = threadIdx.x >> 5;
    const int lane = threadIdx.x & 31;
    __syncthreads();                       // protect LDS scratch reuse
    if (lane == 0) sh[wave] = m;
    __syncthreads();
    float r = sh[0];
#pragma unroll
    for (int i = 1; i < NW; ++i) r = fmaxf(r, sh[i]);   // same-addr reads: LDS broadcast
    return r;
}

// Block-wide (sum, count); 2 barriers; results broadcast to all threads.
__device__ __forceinline__ void block_reduce2(float& s, float& c,
                                              float* shs, float* shc) {
#pragma unroll
    for (int m = 16; m > 0; m >>= 1) {
        s += __shfl_xor(s, m, 32);
        c += __shfl_xor(c, m, 32);
    }
    const int wave = threadIdx.x >> 5;
    const int lane = threadIdx.x & 31;
    __syncthreads();                       // protect LDS scratch reuse
    if (lane == 0) { shs[wave] = s; shc[wave] = c; }
    __syncthreads();
    float ss = 0.f, cc = 0.f;
#pragma unroll
    for (int i = 0; i < NW; ++i) { ss += shs[i]; cc += shc[i]; }
    s = ss;
    c = cc;
}

__global__ void __launch_bounds__(NT)
Sparsemax_80487687127239_kernel(const float* __restrict__ x,
                                float* __restrict__ y,
                                int nrows) {
    __shared__ __align__(16) float buf[2][D];   // double buffer: 2 x 8KB
    __shared__ float shs[NW];
    __shared__ float shc[NW];

    const int t = threadIdx.x;
    const long long row0 = (long long)blockIdx.x * RPB;
    long long rem = (long long)nrows - row0;
    if (rem <= 0) return;
    const int nr = rem < RPB ? (int)rem : RPB;

    // Prime the pipeline: async-stage row 0 into buffer 0.
    {
        const float* src = x + row0 * (long long)D;
        async_copy_b128(src + 4 * t,         &buf[0][4 * t]);
        async_copy_b128(src + D / 2 + 4 * t, &buf[0][D / 2 + 4 * t]);
    }

    for (int r = 0; r < nr; ++r) {
        const int cur = r & 1;

        // Prefetch next row into the other buffer, then wait for current row.
        if (r + 1 < nr) {
            const float* src = x + (row0 + r + 1) * (long long)D;
            async_copy_b128(src + 4 * t,         &buf[cur ^ 1][4 * t]);
            async_copy_b128(src + D / 2 + 4 * t, &buf[cur ^ 1][D / 2 + 4 * t]);
            WAIT_ASYNCCNT(2);   // current row's 2 loads done; prefetch in flight
        } else {
            WAIT_ASYNCCNT(0);
        }
        __syncthreads();        // all waves' portions of buf[cur] visible

        // Pull my 8 elements into registers; row lives in regs from here on.
        v4f a = *(const v4f*)&buf[cur][4 * t];
        v4f b = *(const v4f*)&buf[cur][D / 2 + 4 * t];
        __syncthreads();        // buf[cur] free for refill next iteration's issue

        float v[8];
        v[0] = a.x; v[1] = a.y; v[2] = a.z; v[3] = a.w;
        v[4] = b.x; v[5] = b.y; v[6] = b.z; v[7] = b.w;

        // --- Warm-started Michelot: exact sparsemax threshold, no sort ---
        // tau* >= max-1 always; {x > max-1} is a superset of the true support,
        // so Michelot started at tau0 = max-1 stays monotone and exact, but
        // begins with a support of ~8 elements instead of 2048.
        float lm = v[0];
#pragma unroll
        for (int i = 1; i < 8; ++i) lm = fmaxf(lm, v[i]);
        const float M = block_reduce_max(lm, shs);

        float tau = M - 1.0f;
        float kprev = 0.0f;     // impossible count -> first iteration never breaks

#pragma unroll 1
        for (int it = 0; it < 32; ++it) {
            float sl = 0.f, kl = 0.f;
#pragma unroll
            for (int i = 0; i < 8; ++i) {
                if (v[i] > tau) { sl += v[i]; kl += 1.0f; }
            }
            block_reduce2(sl, kl, shs, shc);   // kl broadcast: uniform branch
            if (kl == kprev) break;            // support stable -> tau exact
            tau = (sl - 1.0f) / kl;
            kprev = kl;
        }

        // --- Emit p = max(x - tau, 0), streaming stores ---
        v4f oa, ob;
        oa.x = fmaxf(v[0] - tau, 0.f); oa.y = fmaxf(v[1] - tau, 0.f);
        oa.z = fmaxf(v[2] - tau, 0.f); oa.w = fmaxf(v[3] - tau, 0.f);
        ob.x = fmaxf(v[4] - tau, 0.f); ob.y = fmaxf(v[5] - tau, 0.f);
        ob.z = fmaxf(v[6] - tau, 0.f); ob.w = fmaxf(v[7] - tau, 0.f);

        float* dst = y + (row0 + r) * (long long)D;
        __builtin_nontemporal_store(oa, (v4f*)(dst + 4 * t));
        __builtin_nontemporal_store(ob, (v4f*)(dst + D / 2 + 4 * t));
    }
}

extern "C" void kernel_launch(void* const* d_in, const int* in_sizes, int n_in,
                              void* d_out, int out_size, void* d_ws, size_t ws_size,
                              hipStream_t stream) {
    const float* x = (const float*)d_in[0];
    float* y = (float*)d_out;
    const long long total = (long long)in_sizes[0];
    const int rows = (int)(total / D);            // 64*1024 = 65536
    const int blocks = (rows + RPB - 1) / RPB;    // 8192
    Sparsemax_80487687127239_kernel<<<blocks, NT, 0, stream>>>(x, y, rows);
}